// cross_entropy_concept_loss_35622458753051
// MI455X (gfx1250) — compile-verified
//
#include <hip/hip_runtime.h>

typedef __attribute__((ext_vector_type(2))) float v2f;
typedef __attribute__((ext_vector_type(4))) float v4f;
typedef __attribute__((ext_vector_type(8))) float v8f;

#define BB    64
#define CC    3
#define FF    112
#define WHX   50176        // 224*224
#define KSEL  5120
#define NTSEL 1024
#define PSORT 8192         // pow2 >= max candidates (count_lt < K, plus ties)

// ---------------------------------------------------------------------------
// Kernel 0: clone batch_X -> d_out. Read-once / write-once stream: use
// non-temporal on both sides so it does not displace the attr map in L2.
// ---------------------------------------------------------------------------
__global__ __launch_bounds__(256) void k_copy(const v4f* __restrict__ src,
                                              v4f* __restrict__ dst, int n4) {
  int i = blockIdx.x * 256 + threadIdx.x;
  if (i < n4) {
    v4f v = __builtin_nontemporal_load(src + i);
    __builtin_nontemporal_store(v, dst + i);
  }
}

// ---------------------------------------------------------------------------
// Kernel 1: attr[b, s] = |sum_f attributions[b,f,s]| / 112 via WMMA f32 16x16x4
// One wave -> 16 consecutive spatial positions of one sample.
// A (16x4) = all ones; B (4x16) tile = 4 f-channels x 16 spatial positions.
// f32 layout: VGPR0 lanes0-15 K=0 / lanes16-31 K=2; VGPR1 K=1 / K=3.
// Exact grid (no tail) so EXEC is all ones for every WMMA.
// Attribution loads are non-temporal: 1.44 GB read-once (7.5x L2) must not
// evict the 12.8 MB attr intermediate that k_select re-reads 5x from L2.
// attr_out stores stay regular-temporal so they remain L2-resident.
// ---------------------------------------------------------------------------
__global__ __launch_bounds__(256) void k_reduce_wmma(const float* __restrict__ attributions,
                                                     float* __restrict__ attr_out) {
  const int wave = (blockIdx.x << 3) | (threadIdx.x >> 5);
  const int lane = threadIdx.x & 31;
  const int half = lane >> 4;        // 0: K=0,1   1: K=2,3
  const int l    = lane & 15;        // N (spatial within tile)
  const int b    = wave / (WHX / 16);
  const int s0   = (wave % (WHX / 16)) << 4;

  const float* base = attributions + (long long)b * FF * WHX + s0 + l;

  v2f a; a.x = 1.0f; a.y = 1.0f;     // ones matrix (all M, all K)
  v8f c = {};

  #pragma unroll 4
  for (int f0 = 0; f0 < FF; f0 += 4) {
    const float* p = base + (long long)(f0 + half * 2) * WHX;
    v2f bt;
    bt.x = __builtin_nontemporal_load(p);
    bt.y = __builtin_nontemporal_load(p + WHX);
    // 8 args: (neg_a, A, neg_b, B, c_mod, C, reuse_a, reuse_b)
    c = __builtin_amdgcn_wmma_f32_16x16x4_f32(false, a, false, bt,
                                              (short)0, c, false, false);
  }
  // D row M=0 lives in c[0] of lanes 0..15 (N = lane)
  if (half == 0)
    attr_out[(long long)b * WHX + s0 + l] = __builtin_fabsf(c[0]) * (1.0f / 112.0f);
}

// ---------------------------------------------------------------------------
// Kernel 2: per-sample exact top-K-smallest with top_k ordering, then scatter.
// One 1024-thread workgroup per sample (64 blocks).
//   1) 4-pass 8-bit radix select on u32 key bits (attr >= 0 -> monotone)
//   2) compact (key <= T) as packed 48-bit (key<<16 | idx) into LDS
//   3) bitonic sort 8192 u64 in LDS (unique keys -> stable order by value,idx)
//   4) scatter random_values for c=0..2 into the cloned output
// ---------------------------------------------------------------------------
__global__ __launch_bounds__(NTSEL) void k_select(const float* __restrict__ attr_ws,
                                                  const float* __restrict__ rv,
                                                  float* __restrict__ out) {
  __shared__ unsigned long long s_buf[PSORT];   // 64 KB
  __shared__ unsigned int s_hist[256];
  __shared__ unsigned int s_prefix, s_remaining, s_count;

  const int b   = blockIdx.x;
  const int tid = threadIdx.x;
  const float* attr = attr_ws + (long long)b * WHX;

  // ---- radix select: exact K-th smallest key ----
  unsigned int prefix = 0, remaining = KSEL;
  for (int pass = 0; pass < 4; ++pass) {
    const int shift = 24 - 8 * pass;
    for (int i = tid; i < 256; i += NTSEL) s_hist[i] = 0;
    __syncthreads();
    for (int i = tid; i < WHX; i += NTSEL) {
      unsigned int key = __float_as_uint(attr[i]);
      bool match = (pass == 0) || ((key >> (shift + 8)) == prefix);
      if (match) atomicAdd(&s_hist[(key >> shift) & 0xFFu], 1u);
    }
    __syncthreads();
    if (tid == 0) {
      unsigned int cum = 0; int d = 0;
      for (; d < 256; ++d) {
        unsigned int h = s_hist[d];
        if (cum + h >= remaining) break;
        cum += h;
      }
      if (d == 256) d = 255;
      s_prefix    = (prefix << 8) | (unsigned int)d;
      s_remaining = remaining - cum;
    }
    __syncthreads();
    prefix = s_prefix; remaining = s_remaining;
    __syncthreads();
  }
  const unsigned int T = prefix;   // exact 32-bit threshold key

  // ---- compact candidates (key <= T) ----
  if (tid == 0) s_count = 0;
  __syncthreads();
  for (int i = tid; i < WHX; i += NTSEL) {
    unsigned int key = __float_as_uint(attr[i]);
    if (key <= T) {
      unsigned int p = atomicAdd(&s_count, 1u);
      if (p < PSORT)
        s_buf[p] = ((unsigned long long)key << 16) | (unsigned long long)(unsigned int)i;
    }
  }
  __syncthreads();
  unsigned int M = s_count; if (M > PSORT) M = PSORT;
  for (int i = tid; i < PSORT; i += NTSEL)
    if ((unsigned int)i >= M) s_buf[i] = ~0ull;   // pad sorts last (real keys < 2^48)
  __syncthreads();

  // ---- bitonic sort (ascending), pairs disjoint per step ----
  for (unsigned int k = 2; k <= PSORT; k <<= 1) {
    for (unsigned int j = k >> 1; j > 0; j >>= 1) {
      for (unsigned int t = tid; t < PSORT; t += NTSEL) {
        unsigned int ixj = t ^ j;
        if (ixj > t) {
          unsigned long long x = s_buf[t], y = s_buf[ixj];
          bool up = ((t & k) == 0);
          if (up ? (x > y) : (x < y)) { s_buf[t] = y; s_buf[ixj] = x; }
        }
      }
      __syncthreads();
    }
  }

  // ---- scatter: first K sorted entries == top_k(-attr) index order ----
  float* outb       = out + (long long)b * CC * WHX;
  const float* rvb  = rv  + (long long)b * CC * KSEL;
  for (int t = tid; t < KSEL; t += NTSEL) {
    unsigned int sidx = (unsigned int)(s_buf[t] & 0xFFFFull);
    outb[0 * WHX + sidx] = rvb[0 * KSEL + t];
    outb[1 * WHX + sidx] = rvb[1 * KSEL + t];
    outb[2 * WHX + sidx] = rvb[2 * KSEL + t];
  }
}

// ---------------------------------------------------------------------------
extern "C" void kernel_launch(void* const* d_in, const int* in_sizes, int n_in,
                              void* d_out, int out_size, void* d_ws, size_t ws_size,
                              hipStream_t stream) {
  const float* batch_X       = (const float*)d_in[0];  // [64,3,224,224]
  const float* attributions  = (const float*)d_in[1];  // [64,112,224,224]
  const float* random_values = (const float*)d_in[2];  // [64,3,5120]
  float* out     = (float*)d_out;
  float* ws_attr = (float*)d_ws;                       // 64*50176 floats = 12.85 MB

  // 1) clone batch_X -> out (must precede the scatter; stream order handles it)
  const int n4 = (BB * CC * WHX) / 4;
  k_copy<<<(n4 + 255) / 256, 256, 0, stream>>>((const v4f*)batch_X,
                                               (v4f*)d_out, n4);

  // 2) F-reduction via WMMA: exact grid, 8 waves/block
  const int nwaves = BB * (WHX / 16);                  // 200704
  k_reduce_wmma<<<nwaves / 8, 256, 0, stream>>>(attributions, ws_attr);

  // 3) per-sample select + sort + scatter
  k_select<<<BB, NTSEL, 0, stream>>>(ws_attr, random_values, out);
}